// CapsuleLayer_59880434040884
// MI455X (gfx1250) — compile-verified
//
#include <hip/hip_runtime.h>
#include <math.h>

// ---------------------------------------------------------------------------
// Capsule dynamic routing, fused for MI455X (gfx1250, wave32).
//   u_hat[b,i,j,k] = sum_d x[b,i,d] * W[i,j,d,k]   (recomputed via WMMA f32)
//   3 routing iterations, u_hat tile kept in LDS between its two uses.
// ---------------------------------------------------------------------------

#define B_TOT    128
#define I_TOT    1152
#define J_TOT    32
#define D_TOT    8
#define K_TOT    16
#define N_TOT    (J_TOT * K_TOT)     // 512, flattened (j,k)
#define ROUTINGS 3

#define BT       16                  // batch tile (WMMA M)
#define IWG      32                  // i's per workgroup
#define ICSUB    4                   // i's per sub-chunk (u_hat LDS slab)
#define NSUB     (IWG / ICSUB)       // 8 sub-chunks
#define THREADS  256                 // 8 waves

// LDS partition (floats)
#define U_FLOATS (ICSUB * BT * N_TOT)   // 32768  (128 KB)
#define S_FLOATS (BT * N_TOT)           // 8192   (32 KB)
#define V_FLOATS (BT * N_TOT)           // 8192   (32 KB)
#define C_FLOATS (BT * ICSUB * J_TOT)   // 2048   (8 KB)
#define SMEM_FLOATS (U_FLOATS + S_FLOATS + V_FLOATS + C_FLOATS)
#define SMEM_BYTES  (SMEM_FLOATS * 4)   // 204800 bytes

typedef __attribute__((ext_vector_type(2))) float v2f;
typedef __attribute__((ext_vector_type(8))) float v8f;

// -------------------------------- init -------------------------------------
__global__ void caps_init_kernel(float* __restrict__ bij, float* __restrict__ s) {
    const int n_b = B_TOT * I_TOT * J_TOT;            // 4718592
    const int n_s = B_TOT * J_TOT * K_TOT;            // 65536
    for (int idx = blockIdx.x * blockDim.x + threadIdx.x; idx < n_b + n_s;
         idx += gridDim.x * blockDim.x) {
        if (idx < n_b) bij[idx] = 0.0f;
        else           s[idx - n_b] = 0.0f;
    }
}

// ------------------------- fused routing iteration --------------------------
// grid: (I_TOT/IWG = 36, B_TOT/BT = 8), block: 256 threads, dynamic LDS.
__global__ __launch_bounds__(THREADS, 1)
void caps_route_kernel(const float* __restrict__ x,   // (128,1152,8)
                       const float* __restrict__ Wm,  // (1152,32,8,16)
                       float* __restrict__ bij,       // (128,1152,32) ws
                       float* __restrict__ s_glob,    // (128,32,16)   ws
                       const float* __restrict__ v_prev, // (128,32,16)
                       int it) {
    extern __shared__ float smem[];
    float* u_lds = smem;                       // [ICSUB][BT][N_TOT]
    float* s_lds = u_lds + U_FLOATS;           // [BT][N_TOT]
    float* v_lds = s_lds + S_FLOATS;           // [BT][N_TOT]
    float* c_lds = v_lds + V_FLOATS;           // [BT][ICSUB][J_TOT]

    const int tid   = threadIdx.x;
    const int lane  = tid & 31;
    const int wv    = tid >> 5;                // wave id 0..7
    const int b0    = blockIdx.y * BT;         // batch base
    const int iblk  = blockIdx.x * IWG;        // i base of workgroup

    // zero s partial, load v of previous iteration
    for (int idx = tid; idx < S_FLOATS; idx += THREADS) s_lds[idx] = 0.0f;
    if (it > 0) {
        for (int idx = tid; idx < V_FLOATS; idx += THREADS) {
            int bl = idx >> 9, n = idx & 511;
            v_lds[idx] = v_prev[(b0 + bl) * N_TOT + n];
        }
    }
    __syncthreads();

    // WMMA operand lane geometry (ISA 7.12.2):
    //  A 16x4 f32: lanes 0-15 -> M=lane, (vgpr0,vgpr1)=(K0,K1); lanes 16-31 -> (K2,K3)
    //  B 4x16 f32: lanes 0-15 -> N=lane, (vgpr0,vgpr1)=(K0,K1); lanes 16-31 -> (K2,K3)
    //  C 16x16 f32: vgpr r: lanes 0-15 -> (M=r,   N=lane)
    //                        lanes 16-31 -> (M=r+8, N=lane-16)
    const int mrow  = (lane < 16) ? 0 : 8;
    const int l16   = lane & 15;
    const int dbase = (lane < 16) ? 0 : 2;
    const int i_loc = wv >> 1;                 // each wave owns one i of the sub-chunk
    const int jbase = (wv & 1) * 16;           // and 16 of the 32 j's

    for (int sc = 0; sc < NSUB; ++sc) {
        const int i_base = iblk + sc * ICSUB;
        const int i_g    = i_base + i_loc;

        // ---- Phase A: u_hat sub-tile via V_WMMA_F32_16X16X4_F32 ------------
        {
            const float* xp = x + ((size_t)(b0 + l16) * I_TOT + i_g) * D_TOT;
            v2f a_lo, a_hi;
            a_lo.x = xp[dbase];     a_lo.y = xp[dbase + 1];
            a_hi.x = xp[dbase + 4]; a_hi.y = xp[dbase + 5];

            #pragma unroll
            for (int t = 0; t < 16; ++t) {
                const int j = jbase + t;                       // n-tile == one j
                const float* wp = Wm + (((size_t)i_g * J_TOT + j) * D_TOT) * K_TOT + l16;
                v2f b_lo, b_hi;
                b_lo.x = wp[(dbase    ) * K_TOT];
                b_lo.y = wp[(dbase + 1) * K_TOT];
                b_hi.x = wp[(dbase + 4) * K_TOT];
                b_hi.y = wp[(dbase + 5) * K_TOT];

                v8f acc = {};
                acc = __builtin_amdgcn_wmma_f32_16x16x4_f32(
                        false, a_lo, false, b_lo, (short)0, acc, false, false);
                acc = __builtin_amdgcn_wmma_f32_16x16x4_f32(
                        false, a_hi, false, b_hi, (short)0, acc, false, false);

                float* up = u_lds + (i_loc * BT) * N_TOT + j * K_TOT + l16;
                #pragma unroll
                for (int r = 0; r < 8; ++r)
                    up[(mrow + r) * N_TOT] = acc[r];
            }
        }
        __syncthreads();

        // ---- Phase B: agreement + b_ij update --------------------------------
        // c_lds[b][i][j] <- b_ij (pre-softmax logits)
        for (int idx = tid; idx < C_FLOATS; idx += THREADS) {
            const int bl = idx >> 7;            // /128
            const int il = (idx >> 5) & 3;
            const int j  = idx & 31;
            float bv;
            if (it == 0) {
                bv = 0.0f;
            } else {
                const float* ur = u_lds + (il * BT + bl) * N_TOT + j * K_TOT;
                const float* vr = v_lds + bl * N_TOT + j * K_TOT;
                float ag = 0.0f;
                #pragma unroll
                for (int k = 0; k < K_TOT; ++k) ag += ur[k] * vr[k];
                const size_t g = ((size_t)(b0 + bl) * I_TOT + (i_base + il)) * J_TOT + j;
                bv = bij[g] + ag;
                if (it + 1 < ROUTINGS) bij[g] = bv;   // persist for next iteration
            }
            c_lds[idx] = bv;
        }
        __syncthreads();

        // ---- Phase C: softmax over j (32) per (b,i) row ----------------------
        if (tid < BT * ICSUB) {
            float* row = c_lds + tid * J_TOT;
            float mx = row[0];
            #pragma unroll
            for (int j = 1; j < J_TOT; ++j) mx = fmaxf(mx, row[j]);
            float sum = 0.0f;
            #pragma unroll
            for (int j = 0; j < J_TOT; ++j) { float e = __expf(row[j] - mx); row[j] = e; sum += e; }
            const float inv = 1.0f / sum;
            #pragma unroll
            for (int j = 0; j < J_TOT; ++j) row[j] *= inv;
        }
        __syncthreads();

        // ---- Phase D: s[b,j,k] += sum_i c[b,i,j]*u_hat[b,i,j,k] -------------
        for (int idx = tid; idx < S_FLOATS; idx += THREADS) {
            const int bl = idx >> 9;
            const int n  = idx & 511;
            const int j  = n >> 4;
            float acc = 0.0f;
            #pragma unroll
            for (int il = 0; il < ICSUB; ++il)
                acc += c_lds[(bl * ICSUB + il) * J_TOT + j] *
                       u_lds[(il * BT + bl) * N_TOT + n];
            s_lds[idx] += acc;
        }
        __syncthreads();   // also protects u_lds reuse next sub-chunk
    }

    // ---- flush s partial to global with fp32 atomics ------------------------
    for (int idx = tid; idx < S_FLOATS; idx += THREADS) {
        const int bl = idx >> 9, n = idx & 511;
        atomicAdd(&s_glob[(b0 + bl) * N_TOT + n], s_lds[idx]);
    }
}

// ------------------------------- squash -------------------------------------
// v[b,j,:] = squash(s[b,j,:]); also re-zeroes s for the next iteration.
__global__ void caps_squash_kernel(float* __restrict__ s, float* __restrict__ v) {
    const int row = blockIdx.x * blockDim.x + threadIdx.x;   // (b,j)
    if (row >= B_TOT * J_TOT) return;
    float* sp = s + row * K_TOT;
    float* vp = v + row * K_TOT;
    float sq = 0.0f;
    float tmp[K_TOT];
    #pragma unroll
    for (int k = 0; k < K_TOT; ++k) { tmp[k] = sp[k]; sq += tmp[k] * tmp[k]; }
    const float scale = (sq / (1.0f + sq)) * rsqrtf(sq + 1e-9f);
    #pragma unroll
    for (int k = 0; k < K_TOT; ++k) { vp[k] = scale * tmp[k]; sp[k] = 0.0f; }
}

// ------------------------------- launch -------------------------------------
extern "C" void kernel_launch(void* const* d_in, const int* in_sizes, int n_in,
                              void* d_out, int out_size, void* d_ws, size_t ws_size,
                              hipStream_t stream) {
    const float* x  = (const float*)d_in[0];   // (128,1152,8)
    const float* Wm = (const float*)d_in[1];   // (1152,32,8,16)
    float* v   = (float*)d_out;                // (128,32,16)
    float* bij = (float*)d_ws;                                     // 18.9 MB
    float* s   = bij + (size_t)B_TOT * I_TOT * J_TOT;              // +256 KB

    caps_init_kernel<<<1024, 256, 0, stream>>>(bij, s);

    dim3 grid(I_TOT / IWG, B_TOT / BT);        // (36, 8)
    for (int it = 0; it < ROUTINGS; ++it) {
        caps_route_kernel<<<grid, THREADS, SMEM_BYTES, stream>>>(x, Wm, bij, s, v, it);
        caps_squash_kernel<<<(B_TOT * J_TOT + 255) / 256, 256, 0, stream>>>(s, v);
    }
}